// GATLayer_73143293051006
// MI455X (gfx1250) — compile-verified
//
#include <hip/hip_runtime.h>
#include <math.h>

typedef __attribute__((ext_vector_type(2))) float v2f;
typedef __attribute__((ext_vector_type(8))) float v8f;

#define D_DIM 128

// ---------------------------------------------------------------------------
// One wave (32 lanes) computes dot(mat[rowBase + m], vec) for m = 0..15 using
// a chain of V_WMMA_F32_16X16X4_F32 ops (exact f32 FMA accumulation).
//
// A-matrix (16x4 f32, ISA 7.12.2): lanes 0-15 hold {K=0,K=1} of row=lane,
// lanes 16-31 hold {K=2,K=3} of row=lane-16  -> per-lane b64 load.
// B-matrix (4x16): column-uniform broadcast of vec[k], same lane/K split.
// D[m][n] then holds the dot product of row m in every column n.
// ---------------------------------------------------------------------------
__device__ __forceinline__ v8f wave_rows16_dot(const float* __restrict__ mat,
                                               const float* __restrict__ vec,
                                               long rowBase, int lane) {
  const int row  = lane & 15;
  const int koff = (lane >> 4) << 1;              // 0 for lanes 0-15, 2 for 16-31
  const float* arow = mat + (rowBase + row) * (long)D_DIM + koff;
  const float* bptr = vec + koff;
  v8f c = {0.f, 0.f, 0.f, 0.f, 0.f, 0.f, 0.f, 0.f};
#pragma unroll
  for (int kb = 0; kb < D_DIM; kb += 4) {
    v2f a = *(const v2f*)(arow + kb);
    v2f b = *(const v2f*)(bptr + kb);
    c = __builtin_amdgcn_wmma_f32_16x16x4_f32(false, a, false, b,
                                              (short)0, c, false, false);
  }
  return c;
}

// C/D layout (16x16 f32): lanes 0-15 VGPR i -> M=i, lanes 16-31 VGPR i -> M=8+i.
// Writer lanes are those with (lane & 8)==0; they own row m = ((lane>>4)<<3)|(lane&7)
// in accumulator element (lane & 7).
__device__ __forceinline__ float v8f_get(const v8f& c, int i) {
  switch (i) {
    case 0: return c[0]; case 1: return c[1]; case 2: return c[2]; case 3: return c[3];
    case 4: return c[4]; case 5: return c[5]; case 6: return c[6]; default: return c[7];
  }
}

// --------------------------- node projections ------------------------------
__global__ void node_proj_kernel(const float* __restrict__ z_src,
                                 const float* __restrict__ z_t,
                                 const float* __restrict__ W,
                                 float* __restrict__ s1,
                                 float* __restrict__ s2,
                                 int nNodes) {
  const int lane = threadIdx.x & 31;
  const int wave = blockIdx.x * (blockDim.x >> 5) + (threadIdx.x >> 5);
  const long nBase = (long)wave * 16;
  if (nBase >= nNodes) return;                    // wave-uniform
  v8f c1 = wave_rows16_dot(z_src, W,          nBase, lane);   // w1 = W[0:128]
  v8f c2 = wave_rows16_dot(z_t,   W + D_DIM,  nBase, lane);   // w2 = W[128:256]
  if ((lane & 8) == 0) {
    const int i = lane & 7;
    const long n = nBase + (((lane >> 4) << 3) | i);
    s1[n] = v8f_get(c1, i);
    s2[n] = v8f_get(c2, i);
  }
}

// --------------------- edge scores (the 328 MB streamer) -------------------
__global__ void edge_score_kernel(const float* __restrict__ z_c,
                                  const float* __restrict__ W,
                                  const float* __restrict__ s1,
                                  const float* __restrict__ s2,
                                  const int* __restrict__ src_idx,
                                  const int* __restrict__ dst_idx,
                                  float* __restrict__ scores,
                                  int nEdges) {
  const int lane = threadIdx.x & 31;
  const int wave = blockIdx.x * (blockDim.x >> 5) + (threadIdx.x >> 5);
  const long eBase = (long)wave * 16;
  if (eBase >= nEdges) return;                    // wave-uniform
  v8f c = wave_rows16_dot(z_c, W + 2 * D_DIM, eBase, lane);   // w3 = W[256:384]
  if ((lane & 8) == 0) {
    const int i = lane & 7;
    const long e = eBase + (((lane >> 4) << 3) | i);
    float sc = v8f_get(c, i) + s1[src_idx[e]] + s2[dst_idx[e]];
    sc = sc > 0.f ? sc : 0.01f * sc;              // leaky_relu(0.01)
    scores[e] = sc;
  }
}

// ------------------------------ softmax pieces -----------------------------
__global__ void seg_init_kernel(float* __restrict__ seg_max,
                                float* __restrict__ seg_sum, int nNodes) {
  int n = blockIdx.x * blockDim.x + threadIdx.x;
  if (n < nNodes) {
    seg_max[n] = -INFINITY;
    seg_sum[n] = 0.f;
  }
}

__global__ void seg_max_kernel(const float* __restrict__ scores,
                               const int* __restrict__ dst_idx,
                               float* __restrict__ seg_max, int nEdges) {
  int e = blockIdx.x * blockDim.x + threadIdx.x;
  if (e >= nEdges) return;
  float v = scores[e];
  float* addr = seg_max + dst_idx[e];
  // monotone bit-pattern trick; valid with -inf initialization
  if (v >= 0.f) atomicMax((int*)addr, __float_as_int(v));
  else          atomicMin((unsigned int*)addr, __float_as_uint(v));
}

__global__ void exp_sum_kernel(const float* __restrict__ scores,
                               const int* __restrict__ dst_idx,
                               const float* __restrict__ seg_max,
                               float* __restrict__ ex,
                               float* __restrict__ seg_sum, int nEdges) {
  int e = blockIdx.x * blockDim.x + threadIdx.x;
  if (e >= nEdges) return;
  int d = dst_idx[e];
  float v = __expf(scores[e] - seg_max[d]);
  ex[e] = v;
  atomicAdd(seg_sum + d, v);
}

__global__ void normalize_kernel(const int* __restrict__ dst_idx,
                                 const float* __restrict__ seg_sum,
                                 float* __restrict__ out, int nEdges) {
  int e = blockIdx.x * blockDim.x + threadIdx.x;
  if (e >= nEdges) return;
  out[e] = out[e] / seg_sum[dst_idx[e]];
}

// ---------------------------------------------------------------------------
extern "C" void kernel_launch(void* const* d_in, const int* in_sizes, int n_in,
                              void* d_out, int out_size, void* d_ws, size_t ws_size,
                              hipStream_t stream) {
  const float* z_src   = (const float*)d_in[0];
  const float* z_t     = (const float*)d_in[1];
  const float* z_c     = (const float*)d_in[2];
  const float* W       = (const float*)d_in[3];
  const int*   src_idx = (const int*)  d_in[4];
  const int*   dst_idx = (const int*)  d_in[5];
  float*       out     = (float*)d_out;

  const int nNodes = in_sizes[0] / D_DIM;   // 10000
  const int nEdges = in_sizes[2] / D_DIM;   // 640000

  float* ws      = (float*)d_ws;
  float* scores  = ws;                        // nEdges
  float* s1      = scores  + nEdges;          // nNodes
  float* s2      = s1      + nNodes;          // nNodes
  float* seg_max = s2      + nNodes;          // nNodes
  float* seg_sum = seg_max + nNodes;          // nNodes

  const int TPB = 256;                        // 8 wave32s per block
  const int wavesPerBlock = TPB / 32;

  // 1. per-node projections s1, s2 (WMMA)
  {
    int waves  = (nNodes + 15) / 16;
    int blocks = (waves + wavesPerBlock - 1) / wavesPerBlock;
    node_proj_kernel<<<blocks, TPB, 0, stream>>>(z_src, z_t, W, s1, s2, nNodes);
  }
  // 2. per-edge scores + leaky relu (WMMA, streams z_c once)
  {
    int waves  = (nEdges + 15) / 16;
    int blocks = (waves + wavesPerBlock - 1) / wavesPerBlock;
    edge_score_kernel<<<blocks, TPB, 0, stream>>>(z_c, W, s1, s2,
                                                  src_idx, dst_idx, scores, nEdges);
  }
  // 3. init segment buffers
  seg_init_kernel<<<(nNodes + TPB - 1) / TPB, TPB, 0, stream>>>(seg_max, seg_sum, nNodes);
  // 4. segment max
  seg_max_kernel<<<(nEdges + TPB - 1) / TPB, TPB, 0, stream>>>(scores, dst_idx,
                                                               seg_max, nEdges);
  // 5. exp + segment sum (ex staged in d_out)
  exp_sum_kernel<<<(nEdges + TPB - 1) / TPB, TPB, 0, stream>>>(scores, dst_idx, seg_max,
                                                               out, seg_sum, nEdges);
  // 6. normalize in place
  normalize_kernel<<<(nEdges + TPB - 1) / TPB, TPB, 0, stream>>>(dst_idx, seg_sum,
                                                                 out, nEdges);
}